// SGConv_41137196761672
// MI455X (gfx1250) — compile-verified
//
#include <hip/hip_runtime.h>

typedef __attribute__((ext_vector_type(2))) float v2f;
typedef __attribute__((ext_vector_type(8))) float v8f;

#define CH 64  // D: in_channels == out_channels

// ---------- degree / normalization ----------
__global__ void k_init_deg(float* __restrict__ deg, int n) {
  int i = blockIdx.x * blockDim.x + threadIdx.x;
  if (i < n) deg[i] = 1.0f;  // self-loop contributes 1 to every node's degree
}

__global__ void k_accum_deg(const int* __restrict__ tgt, float* __restrict__ deg, int nE) {
  int e = blockIdx.x * blockDim.x + threadIdx.x;
  if (e < nE) atomicAdd(&deg[tgt[e]], 1.0f);
}

__global__ void k_dinv(float* __restrict__ deg, int n) {
  int i = blockIdx.x * blockDim.x + threadIdx.x;
  if (i < n) deg[i] = rsqrtf(deg[i]);  // deg >= 1 always (self loops)
}

// ---------- self-loop term: hout[i] = dinv[i]^2 * hin[i] ----------
__global__ void k_self(const float* __restrict__ hin, float* __restrict__ hout,
                       const float* __restrict__ dinv, int n) {
  int idx = blockIdx.x * blockDim.x + threadIdx.x;  // n*32 threads, float2 each
  int i = idx >> 5;
  int p = idx & 31;
  if (i < n) {
    float di = dinv[i];
    float s  = di * di;
    v2f v = *(const v2f*)(hin + (size_t)i * CH + p * 2);
    v2f o;
    o[0] = s * v[0];
    o[1] = s * v[1];
    *(v2f*)(hout + (size_t)i * CH + p * 2) = o;
  }
}

// ---------- edge scatter: one wave32 per edge, 2 channels per lane ----------
__global__ void k_prop(const float* __restrict__ hin, float* __restrict__ hout,
                       const int* __restrict__ src, const int* __restrict__ tgt,
                       const float* __restrict__ dinv, int nE) {
  int wid  = (blockIdx.x * blockDim.x + threadIdx.x) >> 5;  // edge id
  int lane = threadIdx.x & 31;
  if (wid >= nE) return;
  int s = src[wid];                 // broadcast load (same addr across wave)
  int t = tgt[wid];
  float norm = dinv[s] * dinv[t];
  const float* xrow = hin  + (size_t)s * CH + lane * 2;
  float*       orow = hout + (size_t)t * CH + lane * 2;
  v2f v = *(const v2f*)xrow;        // coalesced 256B row read (L2-resident)
  atomicAdd(orow + 0, norm * v[0]); // global_atomic_add_f32, L2 atomic units
  atomicAdd(orow + 1, norm * v[1]);
}

// ---------- final projection: out = h @ W + b via V_WMMA_F32_16X16X4_F32 ----------
// block = 128 threads (4 waves); each wave owns a 16x16 tile: rows
// [blockIdx.x*16, +16), cols [wave*16, +16). 16 WMMA steps cover K=64.
__global__ void k_gemm(const float* __restrict__ h, const float* __restrict__ W,
                       const float* __restrict__ bias, float* __restrict__ out, int n) {
  const int lane  = threadIdx.x & 31;
  const int wave  = threadIdx.x >> 5;
  const int mBase = blockIdx.x * 16;
  if (mBase >= n) return;                    // wave-uniform: EXEC stays all-ones
  const int nBase = wave * 16;
  const int r15   = lane & 15;               // A: M row offset; B: N col offset
  const int kHalf = (lane >> 4) << 1;        // 0 for lanes 0-15, 2 for lanes 16-31

  int mIdx = mBase + r15;
  if (mIdx >= n) mIdx = n - 1;               // safe clamp (n % 16 == 0 here anyway)

  const float* aPtr = h + (size_t)mIdx * CH + kHalf;
  const float* bPtr = W + (size_t)kHalf * CH + (nBase + r15);

  v8f acc = {};
#pragma unroll
  for (int k0 = 0; k0 < CH; k0 += 4) {
    // A fragment: lane L holds A[M=r15][k0+kHalf], A[M=r15][k0+kHalf+1]
    v2f a = *(const v2f*)(aPtr + k0);
    // B fragment: lane L holds W[k0+kHalf][col], W[k0+kHalf+1][col]
    v2f b;
    b[0] = bPtr[(size_t)k0 * CH];
    b[1] = bPtr[(size_t)k0 * CH + CH];
    acc = __builtin_amdgcn_wmma_f32_16x16x4_f32(
        /*neg_a=*/false, a, /*neg_b=*/false, b,
        /*c_mod=*/(short)0, acc, /*reuse_a=*/false, /*reuse_b=*/false);
  }

  const float bv   = bias[nBase + r15];
  const int   mOff = (lane >> 4) * 8;        // C/D: lanes 16-31 hold M = r+8
#pragma unroll
  for (int r = 0; r < 8; ++r) {
    int m = mBase + mOff + r;
    if (m < n) out[(size_t)m * CH + (nBase + r15)] = acc[r] + bv;
  }
}

extern "C" void kernel_launch(void* const* d_in, const int* in_sizes, int n_in,
                              void* d_out, int out_size, void* d_ws, size_t ws_size,
                              hipStream_t stream) {
  const float* x  = (const float*)d_in[0];
  const int*   ei = (const int*)d_in[1];
  const float* W  = (const float*)d_in[2];
  const float* b  = (const float*)d_in[3];
  float*       out = (float*)d_out;

  const int n  = in_sizes[0] / CH;   // 100000 nodes
  const int nE = in_sizes[1] / 2;    // 3200000 edges
  const int* src = ei;               // edge_index[0]
  const int* tgt = ei + nE;          // edge_index[1]

  // workspace layout: dinv [n] | h0 [n*CH] | h1 [n*CH]   (~51.6 MB)
  float* dinv = (float*)d_ws;
  size_t off  = ((size_t)n + 3) & ~(size_t)3;  // 16B align
  float* h0   = dinv + off;
  float* h1   = h0 + (size_t)n * CH;

  const int TB = 256;
  k_init_deg<<<(n + TB - 1) / TB, TB, 0, stream>>>(dinv, n);
  k_accum_deg<<<(nE + TB - 1) / TB, TB, 0, stream>>>(tgt, dinv, nE);
  k_dinv<<<(n + TB - 1) / TB, TB, 0, stream>>>(dinv, n);

  const int selfThreads = n * 32;    // float2 per thread
  const long long propThreads = (long long)nE * 32;  // one wave per edge
  const int propBlocks = (int)((propThreads + TB - 1) / TB);

  // propagation pass 1: x -> h0
  k_self<<<(selfThreads + TB - 1) / TB, TB, 0, stream>>>(x, h0, dinv, n);
  k_prop<<<propBlocks, TB, 0, stream>>>(x, h0, src, tgt, dinv, nE);
  // propagation pass 2: h0 -> h1
  k_self<<<(selfThreads + TB - 1) / TB, TB, 0, stream>>>(h0, h1, dinv, n);
  k_prop<<<propBlocks, TB, 0, stream>>>(h0, h1, src, tgt, dinv, nE);

  // final linear: out = h1 @ W + b (FP32 WMMA)
  k_gemm<<<(n + 15) / 16, 128, 0, stream>>>(h1, W, b, out, n);
}